// SurvivalModel_71399536329172
// MI455X (gfx1250) — compile-verified
//
#include <hip/hip_runtime.h>
#include <math.h>

// Problem constants (from reference setup_inputs)
#define BATCH 8192
#define DD    5
#define EE    4096
#define SS    64
#define NN    32
#define CW    (EE + SS)          // 4160
#define STEEP 64.0f              // 2 * SORTER_SIZE
#define PI_F  3.14159265358979f

typedef __attribute__((ext_vector_type(2))) float v2f;
typedef __attribute__((ext_vector_type(4))) float v4f;
typedef __attribute__((ext_vector_type(8))) float v8f;

// ---------------------------------------------------------------------------
// Kernel 1: per-b fused  normalize -> attention -> decay -> aggregate
// One block (256 threads, 8 waves) per batch row b.
// Embedding row (5 x 4096 f32 = 80KB) is read ONCE from HBM with NON-TEMPORAL
// b128 loads (streamed, no reuse -> don't evict L2), held in regs
// (5 x 4 x v4f per thread).  Element mapping: e = i*1024 + tid*4 + c, so each
// load instruction covers a contiguous 512B per wave (global_load_b128).
// ---------------------------------------------------------------------------
__global__ __launch_bounds__(256) void k_aggregate(
    const float* __restrict__ emb,      // (B, D, E)
    const float* __restrict__ tdiff,    // (B, D)
    const float* __restrict__ sdata,    // (B, S)
    const float* __restrict__ attn_w,   // (1, E)
    const float* __restrict__ attn_b,   // (1,)
    const float* __restrict__ gammap,   // (1,)
    float*       __restrict__ comb,     // (B, CW) workspace (path A) or null
    const float* __restrict__ fc_w,     // (N, CW)  (path B only)
    const float* __restrict__ fc_b,     // (N,)     (path B only)
    float*       __restrict__ risk,     // (B, N)   (path B only)
    int do_gemv)
{
    const int tid  = threadIdx.x;
    const int b    = blockIdx.x;
    const int lane = tid & 31;
    const int wv   = tid >> 5;

    __shared__ float red[8][10];       // per-wave partials: ss[5], dot[5]
    __shared__ float combLds[CW];      // used only in GEMV fallback
    __shared__ float red2[8][NN];      // used only in GEMV fallback

    v4f aw[4];
    v4f xv[DD][4];
    float ss[DD] = {0.f,0.f,0.f,0.f,0.f};
    float dt[DD] = {0.f,0.f,0.f,0.f,0.f};

    const v4f* embRow4 = (const v4f*)(emb + (size_t)b * DD * EE); // 16B aligned
    const v4f* aw4     = (const v4f*)attn_w;

#pragma unroll
    for (int i = 0; i < 4; ++i) aw[i] = aw4[i * 256 + tid];

#pragma unroll
    for (int d = 0; d < DD; ++d) {
#pragma unroll
        for (int i = 0; i < 4; ++i) {
            v4f x = __builtin_nontemporal_load(embRow4 + d * 1024 + i * 256 + tid);
            xv[d][i] = x;
#pragma unroll
            for (int c = 0; c < 4; ++c) {
                ss[d] = fmaf(x[c], x[c], ss[d]);
                dt[d] = fmaf(x[c], aw[i][c], dt[d]);
            }
        }
    }

    // wave32 butterfly reduction
#pragma unroll
    for (int d = 0; d < DD; ++d) {
#pragma unroll
        for (int off = 16; off >= 1; off >>= 1) {
            ss[d] += __shfl_xor(ss[d], off, 32);
            dt[d] += __shfl_xor(dt[d], off, 32);
        }
    }
    if (lane == 0) {
#pragma unroll
        for (int d = 0; d < DD; ++d) { red[wv][d] = ss[d]; red[wv][5 + d] = dt[d]; }
    }
    __syncthreads();

    // every thread finishes the reduction + weight math redundantly (cheap,
    // avoids a broadcast round-trip)
    float coef[DD];
    {
        float ssT[DD] = {0.f,0.f,0.f,0.f,0.f};
        float dtT[DD] = {0.f,0.f,0.f,0.f,0.f};
#pragma unroll
        for (int w = 0; w < 8; ++w)
#pragma unroll
            for (int d = 0; d < DD; ++d) { ssT[d] += red[w][d]; dtT[d] += red[w][5 + d]; }

        const float ab = attn_b[0];
        const float g  = gammap[0];
        float nrm[DD], logit[DD];
        float mx = -1e30f;
#pragma unroll
        for (int d = 0; d < DD; ++d) {
            nrm[d]   = fmaxf(sqrtf(ssT[d]), 1e-12f);
            logit[d] = dtT[d] / nrm[d] + ab;
            mx = fmaxf(mx, logit[d]);
        }
        float sume = 0.f, wgt[DD];
#pragma unroll
        for (int d = 0; d < DD; ++d) { wgt[d] = expf(logit[d] - mx); sume += wgt[d]; }
        float wsum = 0.f;
#pragma unroll
        for (int d = 0; d < DD; ++d) {
            float t     = tdiff[(size_t)b * DD + d];
            float valid = (t != -1.0f) ? 1.0f : 0.0f;
            float decay = expf(-fmaxf(t, 0.0f) * g);
            wgt[d] = (wgt[d] / sume) * decay * valid;
            wsum  += wgt[d];
        }
        wsum = fmaxf(wsum, 1e-8f);
#pragma unroll
        for (int d = 0; d < DD; ++d) coef[d] = (wgt[d] / wsum) / nrm[d];
    }

    if (!do_gemv) {
        // Path A: emit combined row to workspace for the WMMA GEMM kernel.
        // Regular (RT) stores: this 136MB buffer fits in the 192MB L2 and is
        // re-read immediately by k_fc_wmma.
        v4f* crow4 = (v4f*)(comb + (size_t)b * CW);
#pragma unroll
        for (int i = 0; i < 4; ++i) {
            v4f a;
#pragma unroll
            for (int c = 0; c < 4; ++c) {
                float s = 0.f;
#pragma unroll
                for (int d = 0; d < DD; ++d) s = fmaf(xv[d][i][c], coef[d], s);
                a[c] = s;
            }
            crow4[i * 256 + tid] = a;
        }
        if (tid < SS / 4)
            crow4[EE / 4 + tid] = ((const v4f*)(sdata + (size_t)b * SS))[tid];
    } else {
        // Path B (workspace too small): in-block GEMV against fc_w
#pragma unroll
        for (int i = 0; i < 4; ++i) {
            v4f a;
#pragma unroll
            for (int c = 0; c < 4; ++c) {
                float s = 0.f;
#pragma unroll
                for (int d = 0; d < DD; ++d) s = fmaf(xv[d][i][c], coef[d], s);
                a[c] = s;
            }
            ((v4f*)combLds)[i * 256 + tid] = a;
        }
        if (tid < SS / 4)
            ((v4f*)combLds)[EE / 4 + tid] = ((const v4f*)(sdata + (size_t)b * SS))[tid];
        __syncthreads();
        const int n  = tid & 31;
        const int ch = tid >> 5;                 // 8 K-chunks of 520
        const float* wr = fc_w + (size_t)n * CW + ch * 520;
        float p = 0.f;
        for (int k = 0; k < 520; ++k) p = fmaf(wr[k], combLds[ch * 520 + k], p);
        red2[ch][n] = p;
        __syncthreads();
        if (tid < NN) {
            float s = fc_b[tid];
#pragma unroll
            for (int c = 0; c < 8; ++c) s += red2[c][tid];
            risk[(size_t)b * NN + tid] = s;
        }
    }
}

// ---------------------------------------------------------------------------
// Kernel 2: risk = combined(8192x4160) @ fc_w^T(4160x32) + fc_b
// V_WMMA_F32_16X16X4_F32, one wave per 16-row M tile, two N=16 accumulators.
// grid = 64 blocks x 256 threads (8 waves) -> 512 M tiles.
// A reads largely hit L2 (combined was just written and fits in 192MB L2);
// fc_w (532KB) stays L2-resident across all blocks.
// ---------------------------------------------------------------------------
__global__ __launch_bounds__(256) void k_fc_wmma(
    const float* __restrict__ comb,   // (B, CW)
    const float* __restrict__ fc_w,   // (N, CW)
    const float* __restrict__ fc_b,   // (N,)
    float*       __restrict__ risk)   // (B, N)
{
    const int lane  = threadIdx.x & 31;
    const int wv    = threadIdx.x >> 5;
    const int mBase = (blockIdx.x * 8 + wv) * 16;

    const int mrow = mBase + (lane & 15);      // A: lane -> M row
    const int nlo  = lane & 15;                // B: lane -> N col
    const int koff = (lane >> 4) * 2;          // lanes 16-31 hold K+2,K+3

    const float* aptr  = comb + (size_t)mrow * CW + koff;
    const float* b0ptr = fc_w + (size_t)nlo * CW + koff;
    const float* b1ptr = fc_w + (size_t)(nlo + 16) * CW + koff;

    v8f acc0 = {};
    v8f acc1 = {};

#pragma unroll 4
    for (int k = 0; k < CW; k += 4) {
        v2f a  = *(const v2f*)(aptr  + k);
        v2f b0 = *(const v2f*)(b0ptr + k);
        v2f b1 = *(const v2f*)(b1ptr + k);
#if __has_builtin(__builtin_amdgcn_wmma_f32_16x16x4_f32)
        acc0 = __builtin_amdgcn_wmma_f32_16x16x4_f32(false, a, false, b0,
                                                     (short)0, acc0, false, false);
        acc1 = __builtin_amdgcn_wmma_f32_16x16x4_f32(false, a, false, b1,
                                                     (short)0, acc1, false, false);
#else
        // compile-only fallback (layout-incorrect, signals missing builtin)
        acc0[0] = fmaf(a[0], b0[0], acc0[0]);
        acc1[0] = fmaf(a[0], b1[0], acc1[0]);
#endif
    }

    // D layout: VGPR r -> row mBase + r (+8 for lanes 16-31), col = lane&15
    const int col   = lane & 15;
    const int rbase = mBase + ((lane >> 4) ? 8 : 0);
    const float bias0 = fc_b[col];
    const float bias1 = fc_b[col + 16];
#pragma unroll
    for (int r = 0; r < 8; ++r) {
        risk[(size_t)(rbase + r) * NN + col]      = acc0[r] + bias0;
        risk[(size_t)(rbase + r) * NN + col + 16] = acc1[r] + bias1;
    }
}

// ---------------------------------------------------------------------------
// Kernel 3: differentiable bitonic sort (32 elements) -> soft perm matrix.
// One wave32 per batch row. Lane l holds x[l] and column l of X (32 VGPRs).
// For BOTH lanes of a compare pair: new = alpha*own + (1-alpha)*other.
// ---------------------------------------------------------------------------
__global__ __launch_bounds__(256) void k_diffsort(
    const float* __restrict__ risk,   // (B, 32)
    float*       __restrict__ perm)   // (B, 32, 32)
{
    const int lane = threadIdx.x & 31;
    const int b    = blockIdx.x * 8 + (threadIdx.x >> 5);

    float x = risk[(size_t)b * NN + lane];
    float X[NN];
#pragma unroll
    for (int r = 0; r < NN; ++r) X[r] = (r == lane) ? 1.0f : 0.0f;

#pragma unroll
    for (int blk = 0; blk < 5; ++blk) {
#pragma unroll
        for (int lay = 0; lay <= blk; ++lay) {
            const int m = 1 << (blk - lay);
            const float other = __shfl_xor(x, m, 32);
            const bool upper = (lane & m) != 0;
            const bool desc  = ((lane >> (blk + 1)) & 1) != 0;
            const bool isA   = (upper == desc);      // this lane holds position ia
            const float av = isA ? x : other;
            const float bv = isA ? other : x;
            const float alpha = atanf(STEEP * (bv - av)) * (1.0f / PI_F) + 0.5f;
            const float oma   = 1.0f - alpha;
            x = fmaf(alpha, x, oma * other);
#pragma unroll
            for (int r = 0; r < NN; ++r) {
                const float o = __shfl_xor(X[r], m, 32);
                X[r] = fmaf(alpha, X[r], oma * o);
            }
        }
    }

    float* prow = perm + (size_t)b * NN * NN;
#pragma unroll
    for (int r = 0; r < NN; ++r) prow[r * NN + lane] = X[r];   // coalesced
}

// ---------------------------------------------------------------------------
extern "C" void kernel_launch(void* const* d_in, const int* in_sizes, int n_in,
                              void* d_out, int out_size, void* d_ws, size_t ws_size,
                              hipStream_t stream) {
    const float* emb    = (const float*)d_in[0];
    const float* tdiff  = (const float*)d_in[1];
    const float* sdata  = (const float*)d_in[2];
    const float* attn_w = (const float*)d_in[3];
    const float* attn_b = (const float*)d_in[4];
    const float* gamma  = (const float*)d_in[5];
    const float* fc_w   = (const float*)d_in[6];
    const float* fc_b   = (const float*)d_in[7];

    float* out  = (float*)d_out;
    float* risk = out;                          // (8192, 32)
    float* perm = out + (size_t)BATCH * NN;     // (8192, 32, 32)

    const size_t need = (size_t)BATCH * CW * sizeof(float);
    if (ws_size >= need) {
        float* comb = (float*)d_ws;
        k_aggregate<<<BATCH, 256, 0, stream>>>(emb, tdiff, sdata, attn_w, attn_b,
                                               gamma, comb, nullptr, nullptr,
                                               nullptr, 0);
        k_fc_wmma<<<BATCH / 16 / 8, 256, 0, stream>>>(comb, fc_w, fc_b, risk);
    } else {
        k_aggregate<<<BATCH, 256, 0, stream>>>(emb, tdiff, sdata, attn_w, attn_b,
                                               gamma, nullptr, fc_w, fc_b,
                                               risk, 1);
    }
    k_diffsort<<<BATCH / 8, 256, 0, stream>>>(risk, perm);
}